// LearnedScaler_48163763257432
// MI455X (gfx1250) — compile-verified
//
#include <hip/hip_runtime.h>
#include <hip/hip_bf16.h>

typedef unsigned int uint32;
typedef uint32 v4u __attribute__((ext_vector_type(4)));
typedef int    v8i __attribute__((ext_vector_type(8)));
typedef int    v4i __attribute__((ext_vector_type(4)));
typedef float  v2f __attribute__((ext_vector_type(2)));
typedef float  v8f __attribute__((ext_vector_type(8)));

#define NF      2048            // features (columns)
#define NROWS   8192            // observations (rows)
#define CBLK    512             // columns per workgroup
#define TROWS   16              // rows per TDM subtile
#define NSUB    2               // subtiles per workgroup (double-buffered TDM)
#define CHUNK   (TROWS * NSUB)  // 32 rows per workgroup
#define NCHUNK  (NROWS / CHUNK) // 256 row chunks
#define T1      512             // 16 waves per workgroup

// ---------------------------------------------------------------------------
// Build a 2D Tensor-DMA descriptor (D#, ISA cdna5 §8) and issue
// TENSOR_LOAD_TO_LDS. Tile = tile_h rows x tile_w f32 elements, row stride
// row_stride_elems in the source tensor; destination is a compacted row-major
// tile at LDS byte offset lds_off. tensor_dims are set equal to the tile dims
// so no OOB handling engages (caller guarantees in-bounds).
// ---------------------------------------------------------------------------
__device__ __forceinline__ void tdm_load_2d(const void* gptr, uint32 lds_off,
                                            uint32 tile_w, uint32 tile_h,
                                            uint32 row_stride_elems) {
  unsigned long long ga = (unsigned long long)gptr;
  v4u g0;
  g0[0] = 1u;                                   // count=1, user descriptor
  g0[1] = lds_off;                              // lds_addr (bytes)
  g0[2] = (uint32)ga;                           // global_addr[31:0]
  g0[3] = (uint32)((ga >> 32) & 0x01FFFFFFu)    // global_addr[56:32]
        | (2u << 30);                           // type = 2 ("image")
  v8i g1;
  g1[0] = (int)(2u << 16);                      // data_size=2 (4B), no multicast
  g1[1] = (int)((tile_w & 0xFFFFu) << 16);      // tensor_dim0[15:0]
  g1[2] = (int)(((tile_w >> 16) & 0xFFFFu)      // tensor_dim0[31:16]
        | ((tile_h & 0xFFFFu) << 16));          // tensor_dim1[15:0]
  g1[3] = (int)(((tile_h >> 16) & 0xFFFFu)      // tensor_dim1[31:16]
        | ((tile_w & 0xFFFFu) << 16));          // tile_dim0
  g1[4] = (int)(tile_h & 0xFFFFu);              // tile_dim1 (tile_dim2=0)
  g1[5] = (int)row_stride_elems;                // tensor_dim0_stride[31:0]
  g1[6] = 0;                                    // stride hi / dim1_stride lo
  g1[7] = 0;
  v4i z4 = {0, 0, 0, 0};                        // 2D: groups 2/3 unused
  v8i z8 = {0, 0, 0, 0, 0, 0, 0, 0};            // 6-arg toolchain extra group
  __builtin_amdgcn_tensor_load_to_lds(g0, g1, z4, z4, z8, 0);
}

// ---------------------------------------------------------------------------
// Kernel 1: per-(row-chunk, column-block) partial column sum & sum-of-squares.
// TDM streams the tile into LDS; reduction over rows is done with
// V_WMMA_F32_16X16X4_F32 using an all-ones A matrix:
//   D[m,n] = sum_k B[k,n] + C[m,n]   (every row of D = column sums)
// Each wave owns 16 columns per accumulator, 2 column passes -> 32 columns,
// 16 waves -> 512 columns per workgroup.
// ---------------------------------------------------------------------------
__global__ void __launch_bounds__(T1)
welford_stats_partial(const float* __restrict__ x,
                      float* __restrict__ psum,
                      float* __restrict__ psq) {
  extern __shared__ float smem[];               // NSUB * TROWS * CBLK floats (64 KiB)
  const int chunk = blockIdx.x;                 // 0..255
  const int cblk  = blockIdx.y;                 // 0..3
  const int tid   = threadIdx.x;
  const int lane  = tid & 31;
  const int wv    = tid >> 5;

  // Wave 0 issues both tensor DMAs back-to-back (TENSORcnt -> 2), then
  // releases subtile 0 as soon as the first completes (tensorcnt <= 1).
  if (__builtin_amdgcn_readfirstlane(wv) == 0) {
    const int row0 = chunk * CHUNK;
    const int col0 = cblk * CBLK;
    const float* g = x + (size_t)row0 * NF + col0;
    tdm_load_2d(g,                 0u,                        CBLK, TROWS, NF);
    tdm_load_2d(g + (size_t)TROWS * NF,
                (uint32)(TROWS * CBLK * sizeof(float)),       CBLK, TROWS, NF);
    __builtin_amdgcn_s_wait_tensorcnt(1);
  }
  __syncthreads();

  v8f accS0 = {}, accQ0 = {};                   // column pass 0 (cols wv*16 ..)
  v8f accS1 = {}, accQ1 = {};                   // column pass 1 (cols 256+wv*16 ..)
  const v2f ones = {1.0f, 1.0f};                // A(16x4) = all ones
  const int cl   = lane & 15;                   // column within 16-wide group
  const int rsel = (lane >> 4) << 1;            // B striping: upper lanes hold k+2,k+3

  for (int s = 0; s < NSUB; ++s) {
    if (s == 1) {                               // release subtile 1
      if (__builtin_amdgcn_readfirstlane(wv) == 0)
        __builtin_amdgcn_s_wait_tensorcnt(0);
      __syncthreads();
    }
    const float* base = smem + s * (TROWS * CBLK) + wv * 16 + cl;
    for (int r = 0; r < TROWS; r += 4) {
      const float* rowp = base + (r + rsel) * CBLK;
      v2f bA, qA, bB, qB;
      bA.x = rowp[0];          bA.y = rowp[CBLK];        // rows k, k+1 (or k+2,k+3)
      bB.x = rowp[256];        bB.y = rowp[CBLK + 256];  // second column pass
      qA.x = bA.x * bA.x;      qA.y = bA.y * bA.y;
      qB.x = bB.x * bB.x;      qB.y = bB.y * bB.y;
      accS0 = __builtin_amdgcn_wmma_f32_16x16x4_f32(false, ones, false, bA,
                                                    (short)0, accS0, false, false);
      accQ0 = __builtin_amdgcn_wmma_f32_16x16x4_f32(false, ones, false, qA,
                                                    (short)0, accQ0, false, false);
      accS1 = __builtin_amdgcn_wmma_f32_16x16x4_f32(false, ones, false, bB,
                                                    (short)0, accS1, false, false);
      accQ1 = __builtin_amdgcn_wmma_f32_16x16x4_f32(false, ones, false, qB,
                                                    (short)0, accQ1, false, false);
    }
  }

  // D rows are identical; VGPR0 lanes 0..15 hold D[0][n] = colsum(n).
  if (lane < 16) {
    const size_t off = (size_t)chunk * NF + (size_t)cblk * CBLK + wv * 16 + cl;
    psum[off]       = accS0[0];
    psq [off]       = accQ0[0];
    psum[off + 256] = accS1[0];
    psq [off + 256] = accQ1[0];
  }
}

// ---------------------------------------------------------------------------
// Kernel 2: fold 256 partials per column -> mean, invstd = rsqrt(var + eps).
// Welford with N>=2 forgets the initial running stats entirely, so final
// var = (Q - S*mean) / (N-1), mean = S / N.
// ---------------------------------------------------------------------------
__global__ void __launch_bounds__(256)
welford_finalize(const float* __restrict__ psum, const float* __restrict__ psq,
                 float* __restrict__ mean, float* __restrict__ invstd) {
  const int c = blockIdx.x * 256 + threadIdx.x;
  if (c >= NF) return;
  float S = 0.0f, Q = 0.0f;
  for (int k = 0; k < NCHUNK; ++k) {
    S += psum[(size_t)k * NF + c];
    Q += psq [(size_t)k * NF + c];
  }
  const float m   = S * (1.0f / (float)NROWS);
  const float var = (Q - S * m) * (1.0f / (float)(NROWS - 1));
  mean[c]   = m;
  invstd[c] = rsqrtf(var + 1e-6f);
}

// ---------------------------------------------------------------------------
// Kernel 3: out = (x - mean) * invstd, b128-vectorized. x should still be L2
// resident (64 MiB << 192 MB L2) after kernel 1.
// ---------------------------------------------------------------------------
__global__ void __launch_bounds__(256)
welford_normalize(const float4* __restrict__ x4,
                  const float* __restrict__ mean,
                  const float* __restrict__ invstd,
                  float4* __restrict__ out4) {
  const int i  = blockIdx.x * 256 + threadIdx.x;   // 0 .. 8192*2048/4 - 1
  const int cp = (i & (NF / 4 - 1)) << 2;          // starting column of this float4
  const float4 xv = x4[i];
  const float4 m  = *(const float4*)(mean + cp);
  const float4 r  = *(const float4*)(invstd + cp);
  float4 o;
  o.x = (xv.x - m.x) * r.x;
  o.y = (xv.y - m.y) * r.y;
  o.z = (xv.z - m.z) * r.z;
  o.w = (xv.w - m.w) * r.w;
  out4[i] = o;
}

extern "C" void kernel_launch(void* const* d_in, const int* in_sizes, int n_in,
                              void* d_out, int out_size, void* d_ws, size_t ws_size,
                              hipStream_t stream) {
  const float* x = (const float*)d_in[0];
  // d_in[1] (running_mean) and d_in[2] (running_var) are mathematically dead:
  // Welford's n=1 step overwrites mean, and the (n-2)/(n-1) factor zeroes the
  // initial var at n=2.

  float* ws     = (float*)d_ws;
  float* psum   = ws;                               // 256*2048 floats (2 MiB)
  float* psq    = ws + (size_t)NCHUNK * NF;         // 256*2048 floats (2 MiB)
  float* mean   = ws + (size_t)2 * NCHUNK * NF;     // 2048 floats
  float* invstd = mean + NF;                        // 2048 floats

  dim3 g1(NCHUNK, NF / CBLK);                       // 256 x 4 workgroups
  const size_t lds_bytes = (size_t)NSUB * TROWS * CBLK * sizeof(float); // 64 KiB
  welford_stats_partial<<<g1, T1, lds_bytes, stream>>>(x, psum, psq);

  welford_finalize<<<NF / 256, 256, 0, stream>>>(psum, psq, mean, invstd);

  const int n4 = NROWS * NF / 4;                    // 4,194,304 float4s
  welford_normalize<<<n4 / 256, 256, 0, stream>>>((const float4*)x, mean, invstd,
                                                  (float4*)d_out);
}